// GCN_49813030699305
// MI455X (gfx1250) — compile-verified
//
#include <hip/hip_runtime.h>
#include <hip/hip_bf16.h>

typedef __attribute__((ext_vector_type(2))) float v2f;
typedef __attribute__((ext_vector_type(8))) float v8f;

#define DFEAT 128

// ---------------- zero workspace (agg + deg) ----------------
__global__ void GCN_zero_kernel(float* __restrict__ p, long long n) {
    long long i = (long long)blockIdx.x * blockDim.x + threadIdx.x;
    long long stride = (long long)gridDim.x * blockDim.x;
    for (; i < n; i += stride) p[i] = 0.0f;
}

// ---------------- out-degree via f32 atomics ----------------
__global__ void GCN_degree_kernel(const int* __restrict__ src,
                                  float* __restrict__ deg, int E) {
    int i = blockIdx.x * blockDim.x + threadIdx.x;
    if (i < E) atomicAdd(&deg[src[i]], 1.0f);
}

// ---------------- edge scatter-sum: agg[dst] += h[src] ----------------
// One wave (32 lanes) per edge; each lane handles 4 contiguous features.
__global__ __launch_bounds__(256) void GCN_scatter_kernel(
    const float* __restrict__ h, const int* __restrict__ src,
    const int* __restrict__ dst, float* __restrict__ agg, int E) {
    int e = blockIdx.x * 8 + (threadIdx.x >> 5);
    if (e >= E) return;
    int lane = threadIdx.x & 31;
    int s = src[e];
    int d = dst[e];
    float4 v = ((const float4*)(h + (long long)s * DFEAT))[lane];
    float* ap = agg + (long long)d * DFEAT + lane * 4;
    atomicAdd(ap + 0, v.x);
    atomicAdd(ap + 1, v.y);
    atomicAdd(ap + 2, v.z);
    atomicAdd(ap + 3, v.w);
}

// ---------------- out = (norm*agg) @ W * norm + b via V_WMMA_F32_16X16X4_F32 ----
// Block = 256 threads = 8 waves. Block owns a 16-row strip; wave w owns the
// 16x16 output tile at columns [16w, 16w+16). K=128 accumulated in steps of 4.
__global__ __launch_bounds__(256) void GCN_gemm_kernel(
    const float* __restrict__ agg, const float* __restrict__ Wm,
    const float* __restrict__ bias, const float* __restrict__ deg,
    float* __restrict__ out, int N) {
    const int rowBase = blockIdx.x * 16;
    const int lane    = threadIdx.x & 31;
    const int half    = lane >> 4;    // 0: lanes 0-15, 1: lanes 16-31
    const int l15     = lane & 15;
    const int colBase = (threadIdx.x >> 5) * 16;  // wave id * 16

    // A-fragment source row for this lane (rows 0-15 of the strip, duplicated
    // across both lane halves). Clamp for generic N; N=100000 is 16-aligned.
    int aRow = rowBase + l15;
    if (aRow >= N) aRow = N - 1;
    const float dA    = deg[aRow];
    const float normA = (dA > 0.0f) ? (1.0f / sqrtf(dA)) : 0.0f;
    const float* arow = agg + (long long)aRow * DFEAT;

    v8f acc = {0.f, 0.f, 0.f, 0.f, 0.f, 0.f, 0.f, 0.f};

    for (int k0 = 0; k0 < DFEAT; k0 += 4) {
        const int k = k0 + 2 * half;  // this lane's K pair: k, k+1
        // A 16x4 fragment: lane l -> (M = l&15, K = 2*(l>>4)+v), fold in norm.
        v2f a;
        a.x = arow[k]     * normA;
        a.y = arow[k + 1] * normA;
        // B 4x16 fragment: lane l -> (K = 2*(l>>4)+v, N = l&15).
        v2f bb;
        bb.x = Wm[(long long)k       * DFEAT + colBase + l15];
        bb.y = Wm[(long long)(k + 1) * DFEAT + colBase + l15];
        // 8 args: (neg_a, A, neg_b, B, c_mod, C, reuse_a, reuse_b)
        acc = __builtin_amdgcn_wmma_f32_16x16x4_f32(
            false, a, false, bb, (short)0, acc, false, false);
    }

    const float bc = bias[colBase + l15];
#pragma unroll
    for (int r = 0; r < 8; ++r) {
        // D layout: VGPR r, lane l -> (M = r + 8*(l>>4), N = l&15)
        int row = rowBase + r + 8 * half;
        if (row < N) {
            float dd  = deg[row];
            float nrm = (dd > 0.0f) ? (1.0f / sqrtf(dd)) : 0.0f;
            out[(long long)row * DFEAT + colBase + l15] = acc[r] * nrm + bc;
        }
    }
}

extern "C" void kernel_launch(void* const* d_in, const int* in_sizes, int n_in,
                              void* d_out, int out_size, void* d_ws, size_t ws_size,
                              hipStream_t stream) {
    const float* h   = (const float*)d_in[0];
    const float* Wm  = (const float*)d_in[1];
    const float* b   = (const float*)d_in[2];
    const int*   src = (const int*)d_in[3];
    const int*   dst = (const int*)d_in[4];

    const int N = in_sizes[0] / DFEAT;   // 100000
    const int E = in_sizes[3];           // 1700000 (edges + self loops)

    float* out = (float*)d_out;
    float* agg = (float*)d_ws;                     // N*128 f32 = 51.2 MB
    float* deg = agg + (long long)N * DFEAT;       // N f32

    long long zn = (long long)N * (DFEAT + 1);
    GCN_zero_kernel<<<2048, 256, 0, stream>>>(agg, zn);
    GCN_degree_kernel<<<(E + 255) / 256, 256, 0, stream>>>(src, deg, E);
    GCN_scatter_kernel<<<(E + 7) / 8, 256, 0, stream>>>(h, src, dst, agg, E);
    GCN_gemm_kernel<<<(N + 15) / 16, 256, 0, stream>>>(agg, Wm, b, deg, out, N);
}